// MiNbaseNet_15375982920428
// MI455X (gfx1250) — compile-verified
//
#include <hip/hip_runtime.h>
#include <hip/hip_bf16.h>

// Problem constants (match the reference).
#define B_N   1024
#define F_DIM 768
#define M_DIM 512
#define C_N   100
#define PAD_ROWS 2560   // >= sum over classes of ceil16(count_c); worst case 1024 + 100*15 -> 2524

typedef __attribute__((ext_vector_type(2))) float v2f;
typedef __attribute__((ext_vector_type(8))) float v8f;

// ---------------------------------------------------------------------------
// 0) Zero the sorted/padded feature buffer Fs [PAD_ROWS x M_DIM] (f32).
//    Padding rows must be exactly 0 so they contribute nothing to phi/mu.
// ---------------------------------------------------------------------------
__global__ void zero_fs_kernel(float4* __restrict__ fs4) {
    int idx = blockIdx.x * blockDim.x + threadIdx.x;
    const int total4 = PAD_ROWS * M_DIM / 4;
    if (idx < total4) fs4[idx] = make_float4(0.f, 0.f, 0.f, 0.f);
}

// ---------------------------------------------------------------------------
// 1) Meta: class counts, 16-padded prefix offsets, per-sample sorted position.
//    B=1024, C=100 -> trivial serial work, fully deterministic (no atomics).
// ---------------------------------------------------------------------------
__global__ void meta_kernel(const int* __restrict__ labels,
                            int* __restrict__ counts,
                            int* __restrict__ kpad,
                            int* __restrict__ offs,
                            int* __restrict__ posArr) {
    if (threadIdx.x != 0 || blockIdx.x != 0) return;
    for (int c = 0; c < C_N; ++c) counts[c] = 0;
    for (int b = 0; b < B_N; ++b) counts[labels[b]]++;
    int run = 0;
    for (int c = 0; c < C_N; ++c) {
        offs[c] = run;
        int p = (counts[c] + 15) & ~15;   // pad class segment to multiple of 16
        kpad[c] = p;
        run += p;
    }
    int seen[C_N];
    for (int c = 0; c < C_N; ++c) seen[c] = 0;
    for (int b = 0; b < B_N; ++b) {
        int l = labels[b];
        posArr[b] = offs[l] + seen[l]++;
    }
}

// ---------------------------------------------------------------------------
// 2) feat = relu(X @ W), scatter rows into Fs[pos[b]].
//    Block = 256 threads (8 wave32). Block tile: 128(M) x 64(N), K chunks of 16.
//    Wave w owns rows [16w,16w+16), 4 n-subtiles of 16 -> 4x v_wmma_f32_16x16x4_f32
//    per K4-step.
//    LDS: X tile row-padded to 17 (bank-conflict-free stride-17 A reads);
//    W tile stored TRANSPOSED (Wt[col][k], pad 17) so each B fragment
//    {Wt[col][kr], Wt[col][kr+1]} is one ds_load_2addr pair straight into the
//    WMMA source register pair — no repack moves (matches the A-side codegen).
// ---------------------------------------------------------------------------
__global__ __launch_bounds__(256) void gemm_relu_scatter_kernel(
        const float* __restrict__ X, const float* __restrict__ W,
        const int* __restrict__ posArr, float* __restrict__ Fs) {
    __shared__ float Xs[128][17];
    __shared__ float Wt[64][17];   // Wt[col][k]

    const int wave = threadIdx.x >> 5;
    const int lane = threadIdx.x & 31;
    const int half = lane >> 4;      // 0: K={0,1}, 1: K={2,3} per ISA A-layout
    const int l15  = lane & 15;
    const int rowBlk = blockIdx.y * 128;
    const int colBlk = blockIdx.x * 64;

    v8f acc[4] = {};

    for (int k0 = 0; k0 < F_DIM; k0 += 16) {
        // Stage X tile: 128x16 f32 = 512 float4, 2 per thread (coalesced).
        #pragma unroll
        for (int i = 0; i < 2; ++i) {
            int t  = threadIdx.x + i * 256;
            int r  = t >> 2;
            int cq = (t & 3) * 4;
            float4 v = *(const float4*)(X + (size_t)(rowBlk + r) * F_DIM + k0 + cq);
            Xs[r][cq + 0] = v.x; Xs[r][cq + 1] = v.y;
            Xs[r][cq + 2] = v.z; Xs[r][cq + 3] = v.w;
        }
        // Stage W tile transposed: 16x64 f32 = 256 float4, 1 per thread.
        {
            int r  = threadIdx.x >> 4;          // k index 0..15
            int cq = (threadIdx.x & 15) * 4;    // col 0..60
            float4 v = *(const float4*)(W + (size_t)(k0 + r) * M_DIM + colBlk + cq);
            Wt[cq + 0][r] = v.x; Wt[cq + 1][r] = v.y;
            Wt[cq + 2][r] = v.z; Wt[cq + 3][r] = v.w;
        }
        __syncthreads();

        #pragma unroll
        for (int kk = 0; kk < 16; kk += 4) {
            const int kr = kk + half * 2;
            v2f a;
            a.x = Xs[wave * 16 + l15][kr];
            a.y = Xs[wave * 16 + l15][kr + 1];
            #pragma unroll
            for (int nt = 0; nt < 4; ++nt) {
                v2f b;
                b.x = Wt[nt * 16 + l15][kr];
                b.y = Wt[nt * 16 + l15][kr + 1];
                acc[nt] = __builtin_amdgcn_wmma_f32_16x16x4_f32(
                    false, a, false, b, (short)0, acc[nt], false, false);
            }
        }
        __syncthreads();
    }

    // ReLU + scatter-store each output row to its class-sorted slot.
    #pragma unroll
    for (int r = 0; r < 8; ++r) {
        int m   = r + half * 8;                 // C/D layout: VGPR r -> row r / r+8
        int b   = rowBlk + wave * 16 + m;       // sample index
        int pos = posArr[b];
        float* dst = Fs + (size_t)pos * M_DIM + colBlk;
        #pragma unroll
        for (int nt = 0; nt < 4; ++nt)
            dst[nt * 16 + l15] = fmaxf(acc[nt][r], 0.0f);
    }
}

// ---------------------------------------------------------------------------
// 3) phi[c] = Fs_c^T @ Fs_c  (Gram matrix of the class segment, K = kpad[c]).
//    One wave per 16x16 tile of the 512x512 output; 8 tiles per block.
//    A[m][k] = Fs[off+k][m] (transposed read), B[k][n] = Fs[off+k][n]; both
//    fragment loads are 16 contiguous floats per half-wave -> coalesced, and
//    Fs (5 MB) lives in L2 for the whole pass. Output writes (105 MB) are the
//    bandwidth bound. Empty classes (kpad==0) store zeros, matching
//    segment_sum semantics.
// ---------------------------------------------------------------------------
__global__ __launch_bounds__(256) void phi_kernel(
        const float* __restrict__ Fs, const int* __restrict__ offs,
        const int* __restrict__ kpad, float* __restrict__ phi) {
    const int c    = blockIdx.y;
    const int wave = threadIdx.x >> 5;
    const int lane = threadIdx.x & 31;
    const int half = lane >> 4;
    const int l15  = lane & 15;
    const int tile = blockIdx.x * 8 + wave;     // 0..1023
    const int m0   = (tile >> 5) << 4;
    const int n0   = (tile & 31) << 4;
    const int off  = offs[c];
    const int kp   = kpad[c];

    v8f acc = {};
    for (int k0 = 0; k0 < kp; k0 += 4) {
        const float* base = Fs + (size_t)(off + k0 + half * 2) * M_DIM;
        v2f a, b;
        a.x = base[m0 + l15];          // K = k0 + 2*half
        a.y = base[M_DIM + m0 + l15];  // K = k0 + 2*half + 1
        b.x = base[n0 + l15];
        b.y = base[M_DIM + n0 + l15];
        acc = __builtin_amdgcn_wmma_f32_16x16x4_f32(
            false, a, false, b, (short)0, acc, false, false);
    }

    float* out = phi + (size_t)c * M_DIM * M_DIM;
    #pragma unroll
    for (int r = 0; r < 8; ++r) {
        int m = m0 + r + half * 8;
        out[(size_t)m * M_DIM + n0 + l15] = acc[r];
    }
}

// ---------------------------------------------------------------------------
// 4) mu[c] = column sums of the class segment (zero pads are harmless);
//    count[c] written as float. Deterministic fixed-order reduction.
// ---------------------------------------------------------------------------
__global__ void mu_count_kernel(const float* __restrict__ Fs,
                                const int* __restrict__ offs,
                                const int* __restrict__ kpad,
                                const int* __restrict__ counts,
                                float* __restrict__ mu,
                                float* __restrict__ cnt) {
    const int c   = blockIdx.x;
    const int off = offs[c];
    const int kp  = kpad[c];
    for (int col = threadIdx.x; col < M_DIM; col += blockDim.x) {
        float s = 0.f;
        for (int k = 0; k < kp; ++k)
            s += Fs[(size_t)(off + k) * M_DIM + col];
        mu[(size_t)c * M_DIM + col] = s;
    }
    if (threadIdx.x == 0) cnt[c] = (float)counts[c];
}

// ---------------------------------------------------------------------------
extern "C" void kernel_launch(void* const* d_in, const int* in_sizes, int n_in,
                              void* d_out, int out_size, void* d_ws, size_t ws_size,
                              hipStream_t stream) {
    const float* X      = (const float*)d_in[0];   // [1024,768]
    const float* W      = (const float*)d_in[1];   // [768,512]
    const int*   labels = (const int*)d_in[2];     // [1024]

    float* phi = (float*)d_out;                    // [100,512,512]
    float* mu  = phi + (size_t)C_N * M_DIM * M_DIM;
    float* cnt = mu + (size_t)C_N * M_DIM;

    // Workspace layout
    float* Fs     = (float*)d_ws;                          // PAD_ROWS x 512 f32
    int*   posArr = (int*)(Fs + (size_t)PAD_ROWS * M_DIM); // 1024
    int*   counts = posArr + B_N;                          // 100
    int*   kpad   = counts + C_N;                          // 100
    int*   offs   = kpad + C_N;                            // 100

    const int total4 = PAD_ROWS * M_DIM / 4;
    zero_fs_kernel<<<(total4 + 255) / 256, 256, 0, stream>>>((float4*)Fs);
    meta_kernel<<<1, 1, 0, stream>>>(labels, counts, kpad, offs, posArr);
    gemm_relu_scatter_kernel<<<dim3(M_DIM / 64, B_N / 128), 256, 0, stream>>>(
        X, W, posArr, Fs);
    phi_kernel<<<dim3(128, C_N), 256, 0, stream>>>(Fs, offs, kpad, phi);
    mu_count_kernel<<<C_N, 128, 0, stream>>>(Fs, offs, kpad, counts, mu, cnt);
}